// RoIPool_72885595013756
// MI455X (gfx1250) — compile-verified
//
#include <hip/hip_runtime.h>
#include <hip/hip_bf16.h>
#include <stdint.h>

// Problem constants (fixed by the reference module)
#define B_   4
#define C_   256
#define L_   32
#define H_   7
#define W_   7
#define R_   256
#define PL_  4
#define PH_  7
#define PW_  7

#define CH_        8                          // channels staged per block
#define SLICE_     (L_ * H_ * W_)             // 1568 floats per channel
#define TILE_ELEMS (CH_ * SLICE_)             // 12544 floats = 50176 bytes
#define OUT_PER_BLK (CH_ * PL_ * PH_ * PW_)   // 1568 outputs per block (divisible by 4)

typedef unsigned int uint32x4 __attribute__((ext_vector_type(4)));
typedef int          int32x4  __attribute__((ext_vector_type(4)));
typedef int          int32x8  __attribute__((ext_vector_type(8)));

__global__ __launch_bounds__(256) void roipool_tdm_kernel(
    const float* __restrict__ feat,   // [B, C, L, H, W]
    const float* __restrict__ rois,   // [R, 7]
    float* __restrict__ out)          // [R, C, PL, PH, PW]
{
    __shared__ float tile[TILE_ELEMS];

    const int r  = blockIdx.x;
    const int c0 = blockIdx.y * CH_;

    // Per-ROI parameters (block-uniform -> SALU).
    const float* roi = rois + r * 7;
    const int b  = (int)roi[0];
    const int rs = (int)__builtin_rintf(roi[5] * 0.125f);  // round-half-even == jnp.round
    const int re = (int)__builtin_rintf(roi[6] * 0.125f);
    int roi_len  = re - rs + 1; if (roi_len < 1) roi_len = 1;
    const float bs_l = (float)roi_len * 0.25f;             // roi_len / PL, exact in fp32

    // Temporal bin boundaries are block-uniform; lend(pl) == lstart(pl+1) by construction.
    int ls[PL_ + 1];
    #pragma unroll
    for (int p = 0; p <= PL_; ++p) {
        int v = (int)floorf((float)p * bs_l) + rs;
        ls[p] = v < 0 ? 0 : (v > L_ ? L_ : v);
    }

    const float* gsrc = feat + (size_t)(b * C_ + c0) * SLICE_;

    // ---- Stage the (b, c0..c0+7) feature slice into LDS via the Tensor Data Mover ----
#if __has_builtin(__builtin_amdgcn_tensor_load_to_lds)
    if (threadIdx.x < 32) {   // wave 0 only (TDM is a wave-level DMA; EXEC ignored)
        const uint32_t lds_base = (uint32_t)(uintptr_t)(&tile[0]);   // flat low-32 == LDS offset
        const uint64_t ga       = (uint64_t)(uintptr_t)gsrc;

        // D# group 0: count=1 | lds_addr | global_addr | type=2
        uint32x4 g0;
        g0[0] = 1u;                                          // count=1, user descriptor
        g0[1] = lds_base;                                    // lds_addr (bytes)
        g0[2] = (uint32_t)(ga & 0xFFFFFFFFu);                // global_addr[31:0]
        g0[3] = (uint32_t)((ga >> 32) & 0x01FFFFFFu)         // global_addr[56:32]
              | (2u << 30);                                  // type = 2 ("image")

        // D# group 1: data_size=4B, 2D tile 1568 x 8, row stride 1568, no pad/iterate/multicast
        int32x8 g1;
        g1[0] = (int)(2u << 16);                             // wg_mask=0, data_size=2 (4B)
        g1[1] = (int)(((uint32_t)SLICE_ & 0xFFFFu) << 16);   // tensor_dim0[15:0]
        g1[2] = (int)(((uint32_t)SLICE_ >> 16) & 0xFFFFu)    // tensor_dim0[31:16]
              | (int)((uint32_t)CH_ << 16);                  // tensor_dim1[15:0] = 8
        g1[3] = (int)(((uint32_t)SLICE_ & 0xFFFFu) << 16);   // tensor_dim1[31:16]=0, tile_dim0=1568
        g1[4] = (int)((uint32_t)CH_);                        // tile_dim1=8, tile_dim2=0
        g1[5] = (int)SLICE_;                                 // tensor_dim0_stride[31:0] = 1568
        g1[6] = 0;                                           // stride0[47:32]=0, stride1[15:0]=0
        g1[7] = 0;                                           // stride1[47:16]=0

        int32x4 gz; gz[0]=0; gz[1]=0; gz[2]=0; gz[3]=0;      // groups 2/3 unused (<=2D tensor)
#if defined(__clang_major__) && (__clang_major__ >= 23)
        int32x8 gz8; gz8[0]=0; gz8[1]=0; gz8[2]=0; gz8[3]=0; gz8[4]=0; gz8[5]=0; gz8[6]=0; gz8[7]=0;
        __builtin_amdgcn_tensor_load_to_lds(g0, g1, gz, gz, gz8, 0);
#else
        __builtin_amdgcn_tensor_load_to_lds(g0, g1, gz, gz, 0);
#endif
        __builtin_amdgcn_s_wait_tensorcnt(0);
    }
#else
    // Fallback: cooperative coalesced copy
    for (int i = threadIdx.x; i < TILE_ELEMS; i += blockDim.x) tile[i] = gsrc[i];
#endif
    __syncthreads();

    // ---- Temporal max-pool from LDS; 16B-aligned float4 output stores ----
    float* obase = out + (size_t)(r * C_ + c0) * (PL_ * PH_ * PW_);   // dword offset % 4 == 0

    for (int i4 = threadIdx.x * 4; i4 < OUT_PER_BLK; i4 += blockDim.x * 4) {
        float4 vec;
        float* vp = &vec.x;
        #pragma unroll
        for (int e = 0; e < 4; ++e) {
            const int i   = i4 + e;
            const int cl  = i / (PL_ * PH_ * PW_);            // 0..7
            int rem       = i - cl * (PL_ * PH_ * PW_);
            const int pl  = rem / (PH_ * PW_);                // 0..3
            const int sp  = rem - pl * (PH_ * PW_);           // ph*7+pw, 0..48

            const int lstart = ls[pl];
            const int lend   = ls[pl + 1];

            float v = 0.0f;
            if (lend > lstart) {
                const float* src = &tile[cl * SLICE_ + sp];
                float m = -__builtin_inff();
                #pragma unroll 4
                for (int l = lstart; l < lend; ++l)
                    m = fmaxf(m, src[l * (H_ * W_)]);
                v = m;
            }
            vp[e] = v;
        }
        *(float4*)(obase + i4) = vec;
    }
}

extern "C" void kernel_launch(void* const* d_in, const int* in_sizes, int n_in,
                              void* d_out, int out_size, void* d_ws, size_t ws_size,
                              hipStream_t stream) {
    const float* feat = (const float*)d_in[0];   // [4,256,32,7,7] fp32
    const float* rois = (const float*)d_in[1];   // [256,7] fp32
    float* out        = (float*)d_out;           // [256,256,4,7,7] fp32

    dim3 grid(R_, C_ / CH_);                     // 256 x 32 blocks
    roipool_tdm_kernel<<<grid, 256, 0, stream>>>(feat, rois, out);
}